// MyModule_11879879541211
// MI455X (gfx1250) — compile-verified
//
#include <hip/hip_runtime.h>

// Roll-and-insert: out[b, s, :] = cache[b, s+N, :] for s < S-N,
//                  out[b, s, :] = x[b, s-(S-N), :] otherwise.
// Pure streaming copy (zero FLOPs): 128-bit vector loads/stores with
// non-temporal cache policy (256 MiB working set > 192 MB L2, each byte
// touched exactly once). HBM roofline: ~256 MiB / 23.3 TB/s ~= 11 us.
//
// Key codegen point: boundary S-NTOK = 8128 is a multiple of ROWS_PER_BLOCK,
// so every block's 8 rows come entirely from ONE source tensor. Select the
// scalar base pointer once per block; all 8 rows are then base + r*4096 bytes
// (immediate offsets), giving 8 back-to-back b128 loads + 8 b128 stores per
// wave with a single VGPR address component (t*16).

typedef float vfloat4 __attribute__((ext_vector_type(4)));

constexpr int B = 4;
constexpr int S = 8192;
constexpr int D = 1024;
constexpr int NTOK = 64;
constexpr int D4 = D / 4;            // 256 float4 per row == one per lane of a 256-thread block
constexpr int ROWS_PER_BLOCK = 8;    // 8 independent b128 load/store pairs in flight per thread
static_assert((S - NTOK) % ROWS_PER_BLOCK == 0, "blocks must not straddle the boundary");

__global__ __launch_bounds__(256)
void roll_insert_kernel(const vfloat4* __restrict__ x,
                        const vfloat4* __restrict__ cache,
                        vfloat4* __restrict__ out) {
    const int b  = blockIdx.y;
    const int s0 = blockIdx.x * ROWS_PER_BLOCK;
    const int t  = threadIdx.x;            // 0..255 -> one float4 column per lane

    // Single block-uniform source-base select (scalar branch, once per block).
    const vfloat4* __restrict__ src =
        (s0 < S - NTOK)
            ? (cache + ((size_t)b * S    + (size_t)(s0 + NTOK))       * D4)
            : (x     + ((size_t)b * NTOK + (size_t)(s0 - (S - NTOK))) * D4);
    vfloat4* __restrict__ dst = out + ((size_t)b * S + (size_t)s0) * D4;

#pragma unroll
    for (int r = 0; r < ROWS_PER_BLOCK; ++r) {
        // NT policy: stream through without polluting L2 (data never re-read).
        vfloat4 v = __builtin_nontemporal_load(&src[(size_t)r * D4 + t]);
        __builtin_nontemporal_store(v, &dst[(size_t)r * D4 + t]);
    }
}

extern "C" void kernel_launch(void* const* d_in, const int* in_sizes, int n_in,
                              void* d_out, int out_size, void* d_ws, size_t ws_size,
                              hipStream_t stream) {
    const vfloat4* x     = (const vfloat4*)d_in[0];   // [B, 64, D] f32
    const vfloat4* cache = (const vfloat4*)d_in[1];   // [B, S, D]  f32
    vfloat4*       out   = (vfloat4*)d_out;           // [B, S, D]  f32

    dim3 block(256, 1, 1);
    dim3 grid(S / ROWS_PER_BLOCK, B, 1);              // (1024, 4)
    roll_insert_kernel<<<grid, block, 0, stream>>>(x, cache, out);
}